// PearsonLoss_17016660427303
// MI455X (gfx1250) — compile-verified
//
#include <hip/hip_runtime.h>

#define NCH       32
#define CHSTRIDE  65536      // h*w = 256*256
#define THREADS   256

typedef __attribute__((ext_vector_type(2))) float v2f;
typedef __attribute__((ext_vector_type(8))) float v8f;

// Sum 256 f32 values living in LDS using V_WMMA_F32_16X16X4_F32.
// A (16x4 f32) layout: lane L holds (M = L&15, K = 2*(L>>4)) in a.x and K+1 in a.y.
// B = ones(4x16) -> D[i,j] = rowsum_i(A), replicated across N; accumulate 4 chunks.
// C/D layout: VGPR r on lane L holds element (M = r + 8*(L>>4), N = L&15).
// Summing the 8 accumulator VGPRs gives the half-sum; xor-16 shuffle completes it.
// Must be executed by a fully-active wave (EXEC all ones for WMMA).
__device__ __forceinline__ float wmma_sum256(const float* s) {
  const int lane = (int)(threadIdx.x & 31u);
  const int m  = lane & 15;
  const int hi = lane >> 4;
  v2f onesb; onesb.x = 1.0f; onesb.y = 1.0f;
  v8f acc = {};
#pragma unroll
  for (int t = 0; t < 4; ++t) {
    const int idx = t * 64 + m * 4 + 2 * hi;
    v2f a;
    a.x = s[idx];
    a.y = s[idx + 1];
    acc = __builtin_amdgcn_wmma_f32_16x16x4_f32(false, a, false, onesb,
                                                (short)0, acc, false, false);
  }
  float r = ((acc[0] + acc[1]) + (acc[2] + acc[3])) +
            ((acc[4] + acc[5]) + (acc[6] + acc[7]));
  r += __shfl_xor(r, 16, 32);
  return r;
}

__global__ __launch_bounds__(THREADS)
void pearson_pixels(const float* __restrict__ x1, const float* __restrict__ x2,
                    const int* __restrict__ margin, float* __restrict__ block_sums) {
  const int p  = (int)(blockIdx.x * THREADS + threadIdx.x);  // pixel id
  const int b  = p >> 16;          // batch (h*w = 65536)
  const int hw = p & 0xFFFF;
  const size_t base = (size_t)b * (size_t)(NCH * CHSTRIDE) + (size_t)hw;

  // Stream all 32 channels of both inputs into registers (coalesced b32 loads,
  // 64 independent loads per lane -> deep memory-level parallelism).
  float a[NCH], bb[NCH];
#pragma unroll
  for (int c = 0; c < NCH; ++c) {
    a[c]  = x1[base + (size_t)c * CHSTRIDE];
    bb[c] = x2[base + (size_t)c * CHSTRIDE];
  }

  float m1 = a[0], m2 = bb[0];
#pragma unroll
  for (int c = 1; c < NCH; ++c) {
    m1 = fmaxf(m1, a[c]);
    m2 = fmaxf(m2, bb[c]);
  }

  float s1 = 0.f, s2 = 0.f, q1 = 0.f, q2 = 0.f, s12 = 0.f;
#pragma unroll
  for (int c = 0; c < NCH; ++c) {
    const float e1 = __expf(a[c]  - m1);   // v_exp_f32
    const float e2 = __expf(bb[c] - m2);
    s1 += e1;
    s2 += e2;
    q1  = fmaf(e1, e1, q1);
    q2  = fmaf(e2, e2, q2);
    s12 = fmaf(e1, e2, s12);
  }

  // Softmax rows sum to 1 => EX = EY = 1/32 exactly.
  const float inv1 = 1.0f / s1;
  const float inv2 = 1.0f / s2;
  const float kC  = 1.0f / (float)NCH;     // 1/32
  const float kC2 = kC * kC;               // 1/1024 = EX*EY = EX^2 = EY^2
  const float EX2 = q1  * inv1 * inv1 * kC;
  const float EY2 = q2  * inv2 * inv2 * kC;
  const float EXY = s12 * inv1 * inv2 * kC;
  const float num = EXY - kC2;
  const float den = sqrtf(EX2 - kC2) * sqrtf(EY2 - kC2);
  const float score = num / den;
  const float loss  = (margin[p] == 0) ? (1.0f - score) : fmaxf(score, 0.0f);

  __shared__ float sdata[THREADS];
  sdata[threadIdx.x] = loss;
  __syncthreads();
  if (threadIdx.x < 32) {                 // wave 0, fully active -> EXEC all ones
    const float r = wmma_sum256(sdata);
    if ((threadIdx.x & 31u) == 0)
      block_sums[blockIdx.x] = r;
  }
}

__global__ __launch_bounds__(THREADS)
void pearson_finalize(const float* __restrict__ block_sums, int nblocks,
                      float inv_npix, float* __restrict__ out) {
  float s = 0.f;
  for (int i = (int)threadIdx.x; i < nblocks; i += THREADS) s += block_sums[i];
  __shared__ float sdata[THREADS];
  sdata[threadIdx.x] = s;
  __syncthreads();
  if (threadIdx.x < 32) {
    const float r = wmma_sum256(sdata);
    if (threadIdx.x == 0) out[0] = r * inv_npix;
  }
}

extern "C" void kernel_launch(void* const* d_in, const int* in_sizes, int n_in,
                              void* d_out, int out_size, void* d_ws, size_t ws_size,
                              hipStream_t stream) {
  const float* x1     = (const float*)d_in[0];
  const float* x2     = (const float*)d_in[1];
  const int*   margin = (const int*)d_in[2];
  float*       out    = (float*)d_out;
  float*       bsums  = (float*)d_ws;     // nblocks floats of scratch

  const int npix    = in_sizes[2];        // 8*256*256 = 524288 (pixels)
  const int nblocks = npix / THREADS;     // 2048

  pearson_pixels<<<nblocks, THREADS, 0, stream>>>(x1, x2, margin, bsums);
  pearson_finalize<<<1, THREADS, 0, stream>>>(bsums, nblocks,
                                              1.0f / (float)npix, out);
}